// ProgressiveGaussianSplatter_46291157516881
// MI455X (gfx1250) — compile-verified
//
#include <hip/hip_runtime.h>

typedef __attribute__((ext_vector_type(2))) float v2f;
typedef __attribute__((ext_vector_type(8))) float v8f;

#define EPS_F     1e-6f
#define LOG2E_F   1.44269504088896340736f
#define NGAUSS    4096
#define IMG_H     128
#define IMG_W     128
#define GSPLIT    8
#define NG_PER    (NGAUSS / GSPLIT)   /* 512 gaussians per wave */
#define TILES_X   (IMG_W / 16)        /* 8  */
#define NTILES    (IMG_H * TILES_X)   /* 1024 pixel tiles of 16 pixels */
#define NROWS     11                  /* 8 padded feature rows + 3 color rows */

// ---------------------------------------------------------------------------
// Fill output with -1.0 (the reference returns render - 1; d_out is poisoned).
// ---------------------------------------------------------------------------
__global__ void init_out_kernel(float* __restrict__ out, int n) {
    int i = blockIdx.x * blockDim.x + threadIdx.x;
    if (i < n) out[i] = -1.0f;
}

// ---------------------------------------------------------------------------
// Per-gaussian preprocessing: expand (pos, scale, rot) into the quadratic-form
// coefficients of Q(x,y) = a x^2 + b xy + c y^2 + d x + e y + f so the
// per-pixel exponent becomes a [P x 6] * [6 x N] matmul (WMMA-able).
// Coefficients are pre-scaled by log2(e) so the render kernel can use a raw
// v_exp_f32 (exp2) with no per-element multiply. Rows 6,7 are zero padding so
// the B fragment needs no per-lane cndmask. Colors premultiplied by alpha.
// Layout is feature-major: cf[k*NGAUSS + g], k = 0..10.
// ---------------------------------------------------------------------------
__global__ void coeff_kernel(const float* __restrict__ pos,
                             const float* __restrict__ col,
                             const float* __restrict__ lsc,
                             const float* __restrict__ rot,
                             const float* __restrict__ alp,
                             float* __restrict__ cf /* [NROWS][NGAUSS] */) {
    int g = blockIdx.x * blockDim.x + threadIdx.x;
    if (g >= NGAUSS) return;
    float px = pos[2 * g + 0];
    float py = pos[2 * g + 1];
    float sx = __expf(lsc[2 * g + 0]) + EPS_F;
    float sy = __expf(lsc[2 * g + 1]) + EPS_F;
    float isx = 1.0f / (sx * sx);
    float isy = 1.0f / (sy * sy);
    float s, c;
    __sincosf(rot[g], &s, &c);
    // u =  c*x + s*y + tu ; v = -s*x + c*y + tv  (tu,tv fold in the position)
    float tu = -(c * px + s * py);
    float tv =  (s * px - c * py);
    const float k = LOG2E_F;  // fold exp->exp2 conversion into coefficients
    cf[0 * NGAUSS + g] = k * -0.5f * (isx * c * c + isy * s * s);       // x^2
    cf[1 * NGAUSS + g] = k * -((isx - isy) * c * s);                    // x*y
    cf[2 * NGAUSS + g] = k * -0.5f * (isx * s * s + isy * c * c);       // y^2
    cf[3 * NGAUSS + g] = k * -(isx * c * tu - isy * s * tv);            // x
    cf[4 * NGAUSS + g] = k * -(isx * s * tu + isy * c * tv);            // y
    cf[5 * NGAUSS + g] = k * -0.5f * (isx * tu * tu + isy * tv * tv);   // 1
    cf[6 * NGAUSS + g] = 0.0f;                                          // K pad
    cf[7 * NGAUSS + g] = 0.0f;                                          // K pad
    float a = alp[g];
    cf[8 * NGAUSS + g]  = col[3 * g + 0] * a;
    cf[9 * NGAUSS + g]  = col[3 * g + 1] * a;
    cf[10 * NGAUSS + g] = col[3 * g + 2] * a;
}

// ---------------------------------------------------------------------------
// Render: one wave32 owns 16 consecutive pixels of one image row and a slice
// of NG_PER gaussians. Per 16-gaussian tile:
//   Q (16x16) = A(16x8 pixel features) x B(8x16 coeffs)  via 2x WMMA f32 K=4
//   w = exp2(Q)  (raw v_exp_f32; base-2 conversion folded into coeffs)
//   acc[pixel] += w * color[gaussian]  (lane == gaussian column of C layout)
// Epilogue: butterfly-reduce the 16 gaussian lanes per half, atomically add
// split partials into the output image.
// ---------------------------------------------------------------------------
__global__ void __launch_bounds__(256)
render_kernel(const float* __restrict__ cf, float* __restrict__ out) {
    const int tid   = blockIdx.x * blockDim.x + threadIdx.x;
    const int wave  = tid >> 5;
    const int lane  = tid & 31;
    const int tile  = wave & (NTILES - 1);
    const int split = wave / NTILES;

    const int y  = tile / TILES_X;
    const int x0 = (tile % TILES_X) * 16;
    const int m    = lane & 15;   // A: pixel row M ; B/C: gaussian column N
    const int half = lane >> 4;   // lane half selects K pair (A/B), M+8 (C)

    const float px = -1.0f + 2.0f * (float)(x0 + m) / (float)(IMG_W - 1);
    const float py = -1.0f + 2.0f * (float)y / (float)(IMG_H - 1);

    // Pixel feature vector, padded to K=8: [x^2, xy, y^2, x, y, 1, 0, 0]
    const float feat[8] = { px * px, px * py, py * py, px, py, 1.0f, 0.0f, 0.0f };
    // A-matrix 16x4 layout (ISA 7.12.2): VGPR v holds K = v + 2*half
    const v2f a0 = { feat[2 * half + 0], feat[2 * half + 1] };  // K 0..3
    const v2f a1 = { feat[2 * half + 4], feat[2 * half + 5] };  // K 4..7

    float accR[8], accG[8], accB[8];
#pragma unroll
    for (int r = 0; r < 8; ++r) { accR[r] = 0.0f; accG[r] = 0.0f; accB[r] = 0.0f; }

    const int gbase = split * NG_PER;
    for (int t = 0; t < NG_PER; t += 16) {
        const int g = gbase + t + m;  // this lane's gaussian column
        // B-matrix 4x16 chunks, mirrored layout: VGPR v holds K = v + 2*half.
        // Rows 6,7 of cf are physical zeros, so no per-lane select is needed.
        const v2f b0 = { cf[(2 * half + 0) * NGAUSS + g],
                         cf[(2 * half + 1) * NGAUSS + g] };
        const v2f b1 = { cf[(2 * half + 4) * NGAUSS + g],
                         cf[(2 * half + 5) * NGAUSS + g] };

        v8f q = {0.0f, 0.0f, 0.0f, 0.0f, 0.0f, 0.0f, 0.0f, 0.0f};
        q = __builtin_amdgcn_wmma_f32_16x16x4_f32(false, a0, false, b0,
                                                  (short)0, q, false, false);
        q = __builtin_amdgcn_wmma_f32_16x16x4_f32(false, a1, false, b1,
                                                  (short)0, q, false, false);

        const float cr = cf[8 * NGAUSS + g];
        const float cg = cf[9 * NGAUSS + g];
        const float cb = cf[10 * NGAUSS + g];
#pragma unroll
        for (int r = 0; r < 8; ++r) {
            const float w = __builtin_amdgcn_exp2f(q[r]);  // raw v_exp_f32
            accR[r] = fmaf(w, cr, accR[r]);
            accG[r] = fmaf(w, cg, accG[r]);
            accB[r] = fmaf(w, cb, accB[r]);
        }
    }

    // Butterfly reduce over the 16 gaussian-lanes of each half (xor <= 8 stays
    // within a half on wave32); afterwards every lane holds its half's totals
    // for pixel rows (r + 8*half).
#pragma unroll
    for (int mask = 1; mask <= 8; mask <<= 1) {
#pragma unroll
        for (int r = 0; r < 8; ++r) {
            accR[r] += __shfl_xor(accR[r], mask, 32);
            accG[r] += __shfl_xor(accG[r], mask, 32);
            accB[r] += __shfl_xor(accB[r], mask, 32);
        }
    }

    if (m < 8) {
        const int xpix = x0 + m + 8 * half;     // C layout: M = r + 8*half
        const int o = y * IMG_W + xpix;
        atomicAdd(&out[0 * IMG_H * IMG_W + o], accR[m]);
        atomicAdd(&out[1 * IMG_H * IMG_W + o], accG[m]);
        atomicAdd(&out[2 * IMG_H * IMG_W + o], accB[m]);
    }
}

extern "C" void kernel_launch(void* const* d_in, const int* in_sizes, int n_in,
                              void* d_out, int out_size, void* d_ws, size_t ws_size,
                              hipStream_t stream) {
    (void)in_sizes; (void)n_in; (void)ws_size;
    const float* pos = (const float*)d_in[0];  // (1,N,2)
    const float* col = (const float*)d_in[1];  // (1,N,3)
    const float* lsc = (const float*)d_in[2];  // (1,N,2)
    const float* rot = (const float*)d_in[3];  // (1,N)
    const float* alp = (const float*)d_in[4];  // (1,N)
    float* out = (float*)d_out;                // (1,3,H,W) = 49152 floats
    float* cf  = (float*)d_ws;                 // NROWS*NGAUSS floats scratch

    init_out_kernel<<<(out_size + 255) / 256, 256, 0, stream>>>(out, out_size);
    coeff_kernel<<<(NGAUSS + 255) / 256, 256, 0, stream>>>(pos, col, lsc, rot, alp, cf);
    // 1024 tiles * 8 splits = 8192 waves = 1024 blocks of 8 waves
    render_kernel<<<(NTILES * GSPLIT) / 8, 256, 0, stream>>>(cf, out);
}